// SpatialAwareFocalLoss_807453852037
// MI455X (gfx1250) — compile-verified
//
#include <hip/hip_runtime.h>
#include <math.h>

typedef __attribute__((ext_vector_type(2))) float v2f;
typedef __attribute__((ext_vector_type(8))) float v8f;

#define N_TOK      8192
#define NTHREADS   1024
#define PER_THREAD 8

// Spatial-aware focal loss, single-workgroup fused kernel.
// Exploits sorted token_to_line: neighbor windows are contiguous index
// ranges -> prefix sums + binary search instead of the reference's dense
// 8192x8192 mask matvecs. Final reduction via V_WMMA_F32_16X16X4_F32.
__global__ __launch_bounds__(NTHREADS)
void SpatialAwareFocalLoss_kernel(const float* __restrict__ pred,
                                  const float* __restrict__ target,
                                  const int*   __restrict__ lines,
                                  float*       __restrict__ out)
{
    __shared__ int   s_lines[N_TOK];
    __shared__ float s_psig[N_TOK + 1];   // exclusive prefix of sigmoid(pred)
    __shared__ float s_pt  [N_TOK + 1];   // exclusive prefix of target (exact ints in f32)
    __shared__ float s_wsig[32];
    __shared__ float s_wt  [32];
    __shared__ float s_red [32];

    const int tid  = threadIdx.x;
    const int lane = tid & 31;
    const int wid  = tid >> 5;
    const int base = tid * PER_THREAD;

    float sig8[PER_THREAD], t8[PER_THREAD];
    float ls_sig = 0.0f, ls_t = 0.0f;
    float acc = 0.0f;   // focal + penalty accumulator

    // ---- Phase 1: elementwise focal term + local sums, stage lines to LDS ----
    #pragma unroll
    for (int k = 0; k < PER_THREAD; ++k) {
        const int i = base + k;
        const float x = pred[i];
        const float t = target[i];
        s_lines[i] = lines[i];
        const float s = 1.0f / (1.0f + expf(-x));          // sigmoid
        sig8[k] = s; t8[k] = t;
        ls_sig += s; ls_t += t;
        // stable BCE-with-logits
        const float bce = fmaxf(x, 0.0f) - x * t + log1pf(expf(-fabsf(x)));
        const float pt  = expf(-bce);
        const float q   = 1.0f - pt;
        const float q2  = q * q;
        acc += 0.01f * q2 * q2 * bce;                      // ALPHA*(1-pt)^GAMMA*bce
    }

    // ---- Phase 2: block-wide exclusive scan (wave32 shuffle scan + LDS) ----
    float inc_s = ls_sig, inc_t = ls_t;
    #pragma unroll
    for (int off = 1; off < 32; off <<= 1) {
        const float ns = __shfl_up(inc_s, off, 32);
        const float nt = __shfl_up(inc_t, off, 32);
        if (lane >= off) { inc_s += ns; inc_t += nt; }
    }
    if (lane == 31) { s_wsig[wid] = inc_s; s_wt[wid] = inc_t; }
    __syncthreads();

    if (tid < 32) {   // wave 0 scans the 32 wave totals
        const float ws = s_wsig[lane], wt = s_wt[lane];
        float is = ws, it = wt;
        #pragma unroll
        for (int off = 1; off < 32; off <<= 1) {
            const float ns = __shfl_up(is, off, 32);
            const float nt = __shfl_up(it, off, 32);
            if (lane >= off) { is += ns; it += nt; }
        }
        s_wsig[lane] = is - ws;            // exclusive wave offsets
        s_wt[lane]   = it - wt;
        if (lane == 31) { s_psig[N_TOK] = is; s_pt[N_TOK] = it; }
    }
    __syncthreads();

    float run_s = s_wsig[wid] + (inc_s - ls_sig);
    float run_t = s_wt[wid]   + (inc_t - ls_t);
    #pragma unroll
    for (int k = 0; k < PER_THREAD; ++k) {
        const int i = base + k;
        s_psig[i] = run_s; s_pt[i] = run_t;
        run_s += sig8[k];  run_t += t8[k];
    }
    __syncthreads();

    // ---- Phase 3: banded-window penalty via binary search + prefix diffs ----
    #pragma unroll 1
    for (int k = 0; k < PER_THREAD; ++k) {
        const int i  = base + k;
        const int li = s_lines[i];
        // lower bound: first idx with line >= li-2
        int lo = 0, hi = N_TOK;
        const int keyL = li - 2;
        while (lo < hi) { const int mid = (lo + hi) >> 1; if (s_lines[mid] < keyL) lo = mid + 1; else hi = mid; }
        const int lb = lo;
        // upper bound: first idx with line > li+2
        lo = 0; hi = N_TOK;
        const int keyU = li + 2;
        while (lo < hi) { const int mid = (lo + hi) >> 1; if (s_lines[mid] <= keyU) lo = mid + 1; else hi = mid; }
        const int ub = lo;

        const int   cnt  = ub - lb - 1;                       // self excluded
        const float tsum = s_pt[ub] - s_pt[lb] - t8[k];       // exact
        if (cnt > 0 && tsum > 0.0f) {
            const float ssum = s_psig[ub] - s_psig[lb] - sig8[k];
            const float mean = ssum / (float)cnt;             // cnt>0 -> max(cnt,1)==cnt
            acc += 0.3f * (0.1f * mean);                      // SPATIAL_WEIGHT * PENALTY_SCALE
        }
    }

    // ---- Phase 4: reduction; finish with V_WMMA_F32_16X16X4_F32 ----
    #pragma unroll
    for (int off = 16; off > 0; off >>= 1) acc += __shfl_down(acc, off, 32);
    if (lane == 0) s_red[wid] = acc;
    __syncthreads();

    // All 32 waves execute the identical WMMA (EXEC must be all ones).
    // A(16x4): lane layout puts v[lane] at A[lane][0] (lanes 0-15) and
    // A[lane-16][2] (lanes 16-31); B = ones => D[m][n] = v[m] + v[m+16].
    const float v = s_red[lane];
    v2f a; a.x = v;    a.y = 0.0f;
    v2f b; b.x = 1.0f; b.y = 1.0f;
    v8f c = {};
    v8f d = __builtin_amdgcn_wmma_f32_16x16x4_f32(
        /*neg_a=*/false, a, /*neg_b=*/false, b,
        /*c_mod=*/(short)0, c, /*reuse_a=*/false, /*reuse_b=*/false);
    float s = d[0] + d[1] + d[2] + d[3] + d[4] + d[5] + d[6] + d[7];
    s += __shfl_xor(s, 16, 32);          // lane0: rows 0-7+16-23  +  rows 8-15+24-31

    if (tid == 0) out[0] = s * (1.0f / 8192.0f);   // mean over N*C
}

extern "C" void kernel_launch(void* const* d_in, const int* in_sizes, int n_in,
                              void* d_out, int out_size, void* d_ws, size_t ws_size,
                              hipStream_t stream) {
    const float* pred   = (const float*)d_in[0];
    const float* target = (const float*)d_in[1];
    const int*   lines  = (const int*)  d_in[2];
    float*       out    = (float*)d_out;
    (void)in_sizes; (void)n_in; (void)out_size; (void)d_ws; (void)ws_size;

    SpatialAwareFocalLoss_kernel<<<1, NTHREADS, 0, stream>>>(pred, target, lines, out);
}